// EnergyBalanceLoss_47236050321900
// MI455X (gfx1250) — compile-verified
//
#include <hip/hip_runtime.h>
#include <stdint.h>

// ---------------- problem constants ----------------
#define NN   8192      // N nodes
#define KC   64        // clusters
#define BM   32        // rows of E per block
#define BK   64        // reduction (j) tile
#define LDST 68        // padded LDS row stride in floats (64 + 4 pad via TDM)
#define NSTG (NN / BK) // 128 stages

typedef __attribute__((ext_vector_type(2))) float v2f;
typedef __attribute__((ext_vector_type(8))) float v8f;
typedef __attribute__((ext_vector_type(4))) unsigned int v4u;
typedef __attribute__((ext_vector_type(8))) int v8i;
typedef __attribute__((ext_vector_type(4))) int v4i;

// ---------------- TDM 2D tile load (Tensor Data Mover) ----------------
// Builds a D# per CDNA5 ISA and issues tensor_load_to_lds.
// data_size = 4B; LDS padding: 4 dwords every 64 dwords -> LDS row stride 68 floats.
__device__ __forceinline__ void tdm_load_2d(unsigned lds_addr, const void* gptr,
                                            unsigned tensor_w, unsigned tensor_h,
                                            unsigned long long row_stride_elems,
                                            unsigned tile_w, unsigned tile_h) {
  unsigned long long ga = (unsigned long long)(uintptr_t)gptr;
  v4u g0;
  g0.x = 1u;                                             // count=1, user descriptor
  g0.y = lds_addr;                                       // LDS byte address
  g0.z = (unsigned)(ga & 0xffffffffu);                   // global_addr[31:0]
  g0.w = (unsigned)((ga >> 32) & 0x01ffffffu) | (2u << 30); // addr[56:32] | type=2
  v8i g1;
  g1.s0 = (int)((2u << 16)   // data_size = 4 bytes
              | (1u << 20)   // pad_enable
              | (5u << 22)   // pad_interval: every 64 dwords
              | (3u << 25)); // pad_amount: 4 dwords -> stride 68 floats
  g1.s1 = (int)((tensor_w & 0xffffu) << 16);                       // tensor_dim0[15:0]
  g1.s2 = (int)((tensor_w >> 16) | ((tensor_h & 0xffffu) << 16));  // td0 hi | td1 lo
  g1.s3 = (int)((tensor_h >> 16) | (tile_w << 16));                // td1 hi | tile_dim0
  g1.s4 = (int)(tile_h & 0xffffu);                                 // tile_dim1 (tile_dim2=0)
  g1.s5 = (int)(row_stride_elems & 0xffffffffull);                 // tensor_dim0_stride lo
  g1.s6 = (int)((row_stride_elems >> 32) & 0xffffull);             // stride hi (dim1_stride=0)
  g1.s7 = 0;
  v4i z4 = {0, 0, 0, 0};
  v8i z8 = {0, 0, 0, 0, 0, 0, 0, 0};
  __builtin_amdgcn_tensor_load_to_lds(g0, g1, z4, z4, z8, 0);
}

__device__ __forceinline__ float wave_red32(float v) {
  #pragma unroll
  for (int o = 16; o > 0; o >>= 1) v += __shfl_xor(v, o, 32);
  return v;
}

// ---------------- K0: zero workspace ----------------
__global__ void k_zero(float* w, int n) {
  int i = blockIdx.x * blockDim.x + threadIdx.x;
  if (i < n) w[i] = 0.0f;
}

// ---------------- K1: fused streaming pass over E ----------------
// Computes (per block of BM rows): M = E@A strip -> within partial (atomic),
// rowsum (direct store), colsum partials (atomic), all in ONE read of E.
// Double-buffered TDM staging of E tiles (32x64) and A tiles (64x64) in LDS.
__global__ __launch_bounds__(256) void k_fused(const float* __restrict__ E,
                                               const float* __restrict__ A,
                                               float* __restrict__ colsum,
                                               float* __restrict__ rowsum,
                                               float* __restrict__ scal) {
  __shared__ float sE[2][BM * LDST];
  __shared__ float sA[2][BK * LDST];
  __shared__ float sCS[256];

  const int tid  = threadIdx.x;
  const int lane = tid & 31;
  const int wave = tid >> 5;
  const int i0   = blockIdx.x * BM;

  const int rg = wave >> 2; // row tile 0..1 (16 rows each)
  const int cg = wave & 3;  // col tile 0..3 (16 cluster cols each)

  v8f  acc  = {};
  float rAcc = 0.0f;

  const unsigned ldsE0 = (unsigned)(uintptr_t)&sE[0][0];
  const unsigned ldsE1 = (unsigned)(uintptr_t)&sE[1][0];
  const unsigned ldsA0 = (unsigned)(uintptr_t)&sA[0][0];
  const unsigned ldsA1 = (unsigned)(uintptr_t)&sA[1][0];

  if (wave == 0) { // one wave issues DMA (TDM ignores EXEC; per-wave instruction)
    tdm_load_2d(ldsE0, E + (size_t)i0 * NN, NN, NN, NN, BK, BM);
    tdm_load_2d(ldsA0, A, KC, NN, KC, KC, BK);
  }

  for (int s = 0; s < NSTG; ++s) {
    const int p = s & 1;
    if (wave == 0) {
      if (s + 1 < NSTG) {
        const int j1 = (s + 1) * BK;
        tdm_load_2d(p ? ldsE0 : ldsE1, E + (size_t)i0 * NN + j1, NN, NN, NN, BK, BM);
        tdm_load_2d(p ? ldsA0 : ldsA1, A + (size_t)j1 * KC, KC, NN, KC, KC, BK);
        __builtin_amdgcn_s_wait_tensorcnt(2); // stage-s pair done (in-order)
      } else {
        __builtin_amdgcn_s_wait_tensorcnt(0);
      }
    }
    __syncthreads();

    const float* Es = sE[p];
    const float* As = sA[p];

    // --- WMMA f32 16x16x4: 16 k-steps covering BK=64 ---
    const int half = lane >> 4;
    const int l15  = lane & 15;
    #pragma unroll
    for (int kk = 0; kk < 16; ++kk) {
      const int kb = kk * 4 + 2 * half;
      v2f a = *(const v2f*)&Es[(rg * 16 + l15) * LDST + kb]; // A-frag: E rows
      v2f b;
      b.x = As[(kb + 0) * LDST + cg * 16 + l15];             // B-frag: A rows (j)
      b.y = As[(kb + 1) * LDST + cg * 16 + l15];
      acc = __builtin_amdgcn_wmma_f32_16x16x4_f32(false, a, false, b,
                                                  (short)0, acc, false, false);
    }

    // --- row-sum partials: thread covers row tid>>3, 8 cols at (tid&7)*8 ---
    {
      const float* rp = &Es[(tid >> 3) * LDST + (tid & 7) * 8];
      #pragma unroll
      for (int q = 0; q < 8; ++q) rAcc += rp[q];
    }
    // --- col-sum partials: thread covers col tid&63, 8 rows at (tid>>6)*8 ---
    {
      const int c  = tid & 63;
      const int r0 = (tid >> 6) * 8;
      float cp = 0.0f;
      #pragma unroll
      for (int q = 0; q < 8; ++q) cp += Es[(r0 + q) * LDST + c];
      sCS[tid] = cp;
    }
    __syncthreads();
    if (tid < 64) { // waves 0-1: uniform branch
      float v = sCS[tid] + sCS[tid + 64] + sCS[tid + 128] + sCS[tid + 192];
      atomicAdd(&colsum[s * BK + tid], v);
    }
    __syncthreads();
  }

  // finalize rowsum: reduce the 8 lanes sharing a row, direct store (rows owned)
  {
    float v = rAcc;
    v += __shfl_xor(v, 1, 32);
    v += __shfl_xor(v, 2, 32);
    v += __shfl_xor(v, 4, 32);
    if ((tid & 7) == 0) rowsum[i0 + (tid >> 3)] = v;
  }

  // within partial: dot(M-tile, A-tile). C layout: vgpr v, lane l ->
  // row = 16*rg + v + 8*(l>>4), col = 16*cg + (l&15)
  {
    const int half = lane >> 4;
    const int l15  = lane & 15;
    float w = 0.0f;
    #pragma unroll
    for (int v = 0; v < 8; ++v) {
      const int r = i0 + rg * 16 + v + 8 * half;
      const int c = cg * 16 + l15;
      w += acc[v] * A[(size_t)r * KC + c];
    }
    w = wave_red32(w);
    if (lane == 0) atomicAdd(&scal[0], w);
  }
}

// ---------------- K2: argmax + segment counts / position sums ----------------
__global__ __launch_bounds__(256) void k_assign(const float* __restrict__ A,
                                                const float* __restrict__ pos,
                                                float* counts, float* sx, float* sy,
                                                int* maxid, int* ids) {
  int i = blockIdx.x * blockDim.x + threadIdx.x;
  if (i >= NN) return;
  const float* row = A + (size_t)i * KC;
  float best = row[0];
  int bi = 0;
  #pragma unroll 8
  for (int j = 1; j < KC; ++j) {
    float v = row[j];
    if (v > best) { best = v; bi = j; } // first-max like jnp.argmax
  }
  ids[i] = bi;
  atomicAdd(&counts[bi], 1.0f);
  atomicAdd(&sx[bi], pos[2 * i]);
  atomicAdd(&sy[bi], pos[2 * i + 1]);
  atomicMax(maxid, bi);
}

// ---------------- K3: per-node distance + balance partials ----------------
__global__ __launch_bounds__(256) void k_node(const float* __restrict__ pos,
                                              const float* __restrict__ cons,
                                              const float* __restrict__ gen,
                                              const float* __restrict__ colsum,
                                              const float* __restrict__ rowsum,
                                              const float* counts, const float* sx,
                                              const float* sy, const int* ids,
                                              float* dsum, float* scal) {
  int i = blockIdx.x * blockDim.x + threadIdx.x;
  float imb2 = 0.0f, te = 0.0f;
  if (i < NN) {
    int k = ids[i];
    float cnt = counts[k];
    float cx = sx[k] / (cnt + 1e-6f);
    float cy = sy[k] / (cnt + 1e-6f);
    float dx = pos[2 * i] - cx;
    float dy = pos[2 * i + 1] - cy;
    atomicAdd(&dsum[k], sqrtf(dx * dx + dy * dy));
    float imb = (cons[i] - gen[i]) - (colsum[i] - rowsum[i]);
    imb2 = imb * imb;
    te = colsum[i]; // sum of colsum == sum(E)
  }
  imb2 = wave_red32(imb2);
  te = wave_red32(te);
  if ((threadIdx.x & 31) == 0) {
    atomicAdd(&scal[1], imb2);
    atomicAdd(&scal[2], te);
  }
}

// ---------------- K4: final combine ----------------
__global__ void k_final(const float* counts, const float* dsum, const int* maxid,
                        const float* scal, float* out) {
  if (threadIdx.x == 0 && blockIdx.x == 0) {
    float tot = 0.0f;
    for (int k = 0; k < KC; ++k) {
      float c = counts[k];
      if (c >= 2.0f) tot += dsum[k] / (c + 1e-6f);
    }
    float numc = (float)(*maxid) + 1.0f;
    float spatial = tot / (numc + 1e-6f);
    float balance = scal[1] / (float)NN;            // mean(imbalance^2)
    float sizeE = (float)NN * (float)NN;
    float clustering = (scal[2] - 2.0f * scal[0]) / (sizeE + 1e-6f);
    float total = 1.0f * balance + 0.5f * spatial + 0.3f * clustering;
    out[0] = total;
    out[1] = balance;
    out[2] = spatial;
    out[3] = clustering;
  }
}

// ---------------- launch ----------------
// Workspace layout (floats):
//   [0..8191]        colsum
//   [8192..16383]    rowsum
//   [16384]          within   [16385] balSum   [16386] totalE
//   [16387]          maxid (int)
//   [16388..16451]   counts   [16452..16515] sx   [16516..16579] sy
//   [16580..16643]   dsum
//   [16644..24835]   ids (int)
extern "C" void kernel_launch(void* const* d_in, const int* in_sizes, int n_in,
                              void* d_out, int out_size, void* d_ws, size_t ws_size,
                              hipStream_t stream) {
  (void)in_sizes; (void)n_in; (void)out_size; (void)ws_size;
  const float* E    = (const float*)d_in[0];
  const float* A    = (const float*)d_in[1];
  const float* pos  = (const float*)d_in[2];
  const float* cons = (const float*)d_in[3];
  const float* gen  = (const float*)d_in[4];
  float* out = (float*)d_out;

  float* w      = (float*)d_ws;
  float* colsum = w;
  float* rowsum = w + 8192;
  float* scal   = w + 16384;
  int*   maxid  = (int*)(w + 16387);
  float* counts = w + 16388;
  float* sx     = w + 16452;
  float* sy     = w + 16516;
  float* dsum   = w + 16580;
  int*   ids    = (int*)(w + 16644);

  const int zn = 16644;
  k_zero<<<(zn + 255) / 256, 256, 0, stream>>>(w, zn);
  k_fused<<<NN / BM, 256, 0, stream>>>(E, A, colsum, rowsum, scal);
  k_assign<<<NN / 256, 256, 0, stream>>>(A, pos, counts, sx, sy, maxid, ids);
  k_node<<<NN / 256, 256, 0, stream>>>(pos, cons, gen, colsum, rowsum,
                                       counts, sx, sy, ids, dsum, scal);
  k_final<<<1, 64, 0, stream>>>(counts, dsum, maxid, scal, out);
}